// SimpleSSM_31679678775738
// MI455X (gfx1250) — compile-verified
//
#include <hip/hip_runtime.h>
#include <hip/hip_bf16.h>
#include <math.h>

// SimpleSSM: y = hs @ C_w^T + D*x, hs = scan(A_bar, Bu), Bu = x @ B_w^T
// B=8, N=8192, Dm=512, Ds=16. f32 throughout (matches reference precision).
// Memory-bound (~290-420 MB at 23.3 TB/s => ~12-18 us). GEMMs use
// V_WMMA_F32_16X16X4_F32; scan parallelized as a 3-phase chunked linear scan.

#define BSZ    8
#define NSEQ   8192
#define DM     512
#define DS     16
#define CHUNK  128
#define NCHUNK (NSEQ / CHUNK)   // 64
#define HSTR   18               // padded LDS row stride (floats): even => 8B-aligned v2f

typedef __attribute__((ext_vector_type(2))) float v2f;
typedef __attribute__((ext_vector_type(8))) float v8f;

__device__ __forceinline__ float neg_clamped_A(const float* A_log, int s) {
    // a = -clamp(exp(A_log[s]), 1e-8, 10)
    return -fminf(fmaxf(__expf(A_log[s]), 1e-8f), 10.0f);
}

// ---------------------------------------------------------------------------
// Kernel 1: per (b, chunk): Bu = x @ B_w^T via f32 WMMA, then local scan (h0=0)
// in LDS. Writes local states to scratch and chunk-final state.
// grid = B*NCHUNK blocks, 256 threads (8 waves; wave w owns t-tile w).
// ---------------------------------------------------------------------------
__global__ __launch_bounds__(256) void k1_bu_localscan(
    const float* __restrict__ x, const float* __restrict__ A_log,
    const float* __restrict__ B_w, float* __restrict__ hs_local,
    float* __restrict__ chunk_final)
{
    __shared__ float bu[CHUNK * HSTR];

    const int blk  = blockIdx.x;
    const int b    = blk / NCHUNK;
    const int c    = blk % NCHUNK;
    const int tid  = threadIdx.x;
    const int lane = tid & 31;
    const int wave = tid >> 5;
    const int lo   = lane & 15;     // column / row-within-half
    const int hi   = lane >> 4;     // which K pair (A) / K half (B), M half (C)

    // ---- GEMM1: 16(t) x 16(s) tile, K over Dm=512 in steps of 4 ----
    const int    t0   = c * CHUNK + wave * 16;          // global n of tile row 0
    const float* xrow = x   + ((size_t)b * NSEQ + t0 + lo) * DM;  // A: row M=lo
    const float* brow = B_w + (size_t)lo * DM;                    // B: col N=lo

    v8f acc = {};
    #pragma unroll 8
    for (int k = 0; k < DM; k += 4) {
        // A (16x4 f32): lane holds {x[row][k+2*hi], x[row][k+2*hi+1]}
        v2f a  = *(const v2f*)(xrow + k + 2 * hi);
        // B (4x16 f32): lane holds {B_w[s][k+2*hi], B_w[s][k+2*hi+1]} (B = B_w^T)
        v2f bb = *(const v2f*)(brow + k + 2 * hi);
        acc = __builtin_amdgcn_wmma_f32_16x16x4_f32(
            false, a, false, bb, (short)0, acc, false, false);
    }

    // C/D layout: VGPR j -> row M = j + 8*hi, col N = lo
    #pragma unroll
    for (int j = 0; j < 8; ++j) {
        int tl = wave * 16 + j + 8 * hi;               // local t within chunk
        bu[tl * HSTR + lo] = acc[j];
    }
    __syncthreads();

    // ---- local scan: 16 lanes (one per state channel s), h0 = 0 ----
    if (tid < DS) {
        const int s   = tid;
        const float a = neg_clamped_A(A_log, s);
        const float ab = __expf(a * (1.0f / (float)NSEQ));   // A_bar
        float h = 0.0f;
        for (int t = 0; t < CHUNK; ++t) {
            h = ab * h + bu[t * HSTR + s];
            bu[t * HSTR + s] = h;                            // local hs
        }
        chunk_final[((size_t)b * NCHUNK + c) * DS + s] = h;
    }
    __syncthreads();

    // ---- spill local hs chunk to scratch (coalesced) ----
    for (int e = tid; e < CHUNK * DS; e += 256) {
        int t = e >> 4, s = e & 15;
        hs_local[(((size_t)b * NSEQ) + (size_t)c * CHUNK + t) * DS + s] =
            bu[t * HSTR + s];
    }
}

// ---------------------------------------------------------------------------
// Kernel 2: sequential combine across chunks per (b, s) channel.
// chunk_prefix[b][c][s] = state entering chunk c. 1 block, 128 threads.
// ---------------------------------------------------------------------------
__global__ __launch_bounds__(128) void k2_prefix(
    const float* __restrict__ A_log,
    const float* __restrict__ chunk_final,
    float* __restrict__ chunk_prefix)
{
    const int tid = threadIdx.x;
    if (tid >= BSZ * DS) return;
    const int b = tid / DS, s = tid % DS;
    const float a   = neg_clamped_A(A_log, s);
    const float abL = __expf(a * ((float)CHUNK / (float)NSEQ)); // A_bar^CHUNK, exact
    float carry = 0.0f;
    for (int c = 0; c < NCHUNK; ++c) {
        const size_t idx = ((size_t)b * NCHUNK + c) * DS + s;
        chunk_prefix[idx] = carry;
        carry = abL * carry + chunk_final[idx];
    }
}

// ---------------------------------------------------------------------------
// Kernel 3: per (b, chunk): correct local states with incoming chunk state,
// then y = hs @ C_w^T + D*x via f32 WMMA. grid = B*NCHUNK, 256 threads.
// ---------------------------------------------------------------------------
__global__ __launch_bounds__(256) void k3_adjust_gemm2(
    const float* __restrict__ x, const float* __restrict__ A_log,
    const float* __restrict__ C_w, const float* __restrict__ Dvec,
    const float* __restrict__ hs_local, const float* __restrict__ chunk_prefix,
    float* __restrict__ y)
{
    __shared__ float hs[CHUNK * HSTR];

    const int blk  = blockIdx.x;
    const int b    = blk / NCHUNK;
    const int c    = blk % NCHUNK;
    const int tid  = threadIdx.x;
    const int lane = tid & 31;
    const int wave = tid >> 5;
    const int lo   = lane & 15;
    const int hi   = lane >> 4;

    // ---- load local hs + correction: h_t = hl_t + A_bar^(t+1) * H_prev ----
    for (int e = tid; e < CHUNK * DS; e += 256) {
        int t = e >> 4, s = e & 15;
        const float a    = neg_clamped_A(A_log, s);
        const float powv = __expf(a * ((float)(t + 1) / (float)NSEQ)); // exact A_bar^(t+1)
        const float Hp   = chunk_prefix[((size_t)b * NCHUNK + c) * DS + s];
        const float v    = hs_local[(((size_t)b * NSEQ) + (size_t)c * CHUNK + t) * DS + s];
        hs[t * HSTR + s] = v + powv * Hp;
    }
    __syncthreads();

    // ---- GEMM2: wave owns t-tile; A operand (hs 16x16, K=Ds) hoisted ----
    const int tl0 = wave * 16;                  // local row base
    const int tg0 = c * CHUNK + tl0;            // global n base
    v2f aop[4];
    #pragma unroll
    for (int kk = 0; kk < 4; ++kk)
        aop[kk] = *(const v2f*)(&hs[(tl0 + lo) * HSTR + kk * 4 + 2 * hi]);

    for (int d0 = 0; d0 < DM; d0 += 16) {
        v8f acc = {};
        #pragma unroll
        for (int kk = 0; kk < 4; ++kk) {
            // B (16x16 slice of C_w^T): lane holds {C_w[d][kk*4+2hi], C_w[d][kk*4+2hi+1]}
            v2f bb = *(const v2f*)(C_w + (size_t)(d0 + lo) * DS + kk * 4 + 2 * hi);
            acc = __builtin_amdgcn_wmma_f32_16x16x4_f32(
                false, aop[kk], false, bb, (short)0, acc, false, false);
        }
        const float dv = Dvec[d0 + lo];
        #pragma unroll
        for (int j = 0; j < 8; ++j) {
            const int tg = tg0 + j + 8 * hi;
            const size_t off = ((size_t)b * NSEQ + tg) * DM + d0 + lo;
            y[off] = acc[j] + dv * x[off];      // fused D*x epilogue (x from L2)
        }
    }
}

// ---------------------------------------------------------------------------
extern "C" void kernel_launch(void* const* d_in, const int* in_sizes, int n_in,
                              void* d_out, int out_size, void* d_ws, size_t ws_size,
                              hipStream_t stream) {
    const float* x     = (const float*)d_in[0];
    const float* A_log = (const float*)d_in[1];
    const float* B_w   = (const float*)d_in[2];
    const float* C_w   = (const float*)d_in[3];
    const float* Dvec  = (const float*)d_in[4];
    float* y = (float*)d_out;

    // workspace: hs_local (4 MB) | chunk_final (32 KB) | chunk_prefix (32 KB)
    char* ws = (char*)d_ws;
    float* hs_local     = (float*)(ws);
    float* chunk_final  = (float*)(ws + (size_t)BSZ * NSEQ * DS * sizeof(float));
    float* chunk_prefix = (float*)(ws + (size_t)BSZ * NSEQ * DS * sizeof(float)
                                      + (size_t)BSZ * NCHUNK * DS * sizeof(float));

    const int nblocks = BSZ * NCHUNK;   // 512
    k1_bu_localscan<<<nblocks, 256, 0, stream>>>(x, A_log, B_w, hs_local, chunk_final);
    k2_prefix<<<1, 128, 0, stream>>>(A_log, chunk_final, chunk_prefix);
    k3_adjust_gemm2<<<nblocks, 256, 0, stream>>>(x, A_log, C_w, Dvec,
                                                 hs_local, chunk_prefix, y);
}